// DF11Embedding_50422916055142
// MI455X (gfx1250) — compile-verified
//
#include <hip/hip_runtime.h>
#include <stdint.h>

// Embedding gather: out[b] = weight[ids[b]], rows of DIM f32.
// Pure bandwidth problem (~512 MB traffic -> ~22us floor @ 23.3 TB/s).
// CDNA5 async DMA path: global -> LDS -> global with b128 transfers
// (ASYNCcnt), no VGPR round-trip. Output stores are non-temporal so the
// streaming 256 MB write doesn't evict repeated weight rows from the
// 192 MB L2 (~27% of gathered rows are duplicates at this vocab/seq size).

#define DIM        2048
#define ROW_BYTES  (DIM * 4)        // 8192 bytes per embedding row
#define ROWS_PER_BLOCK 2

__global__ __launch_bounds__(256)
void embed_gather_async(const int* __restrict__ ids,
                        const float* __restrict__ weight,
                        float* __restrict__ out,
                        int n_tok)
{
    const uint32_t row0 = (uint32_t)blockIdx.x * ROWS_PER_BLOCK;
    // Branch-free tail: for odd n_tok the last block writes the final row
    // twice with identical data (deterministic, benign). For even n_tok
    // (the actual workload: 32768 tokens) the clamp is a no-op.
    const uint32_t last = (uint32_t)(n_tok - 1);
    const uint32_t row1 = (row0 + 1 < last) ? (row0 + 1) : last;

    // Block-uniform token indices -> scalar loads (KMcnt path).
    const uint32_t tok0 = (uint32_t)ids[row0];
    const uint32_t tok1 = (uint32_t)ids[row1];

    const uint32_t byte_off = threadIdx.x * 32u;  // 32 B/thread, 8 KB/row

    // GVS addressing: uniform 64-bit SGPR base + per-lane 32-bit byte offset.
    // Max src_off ~= 412 MB, max dst_off ~= 268 MB: both < 2^31 (safe i32).
    const uint32_t src0 = tok0 * (uint32_t)ROW_BYTES + byte_off;
    const uint32_t src1 = tok1 * (uint32_t)ROW_BYTES + byte_off;
    const uint32_t dst0 = row0 * (uint32_t)ROW_BYTES + byte_off;
    const uint32_t dst1 = row1 * (uint32_t)ROW_BYTES + byte_off;
    const uint32_t lds0 = byte_off;               // dynamic-LDS base offset 0
    const uint32_t lds1 = byte_off + (uint32_t)ROW_BYTES;

    // ---- global -> LDS: 4 outstanding 128-bit async DMA loads per lane ----
    asm volatile(
        "global_load_async_to_lds_b128 %0, %2, %4\n\t"
        "global_load_async_to_lds_b128 %0, %2, %4 offset:16\n\t"
        "global_load_async_to_lds_b128 %1, %3, %4\n\t"
        "global_load_async_to_lds_b128 %1, %3, %4 offset:16"
        :
        : "v"(lds0), "v"(lds1), "v"(src0), "v"(src1), "s"(weight)
        : "memory");

    // Each wave only reads back LDS bytes its own lanes wrote -> a per-wave
    // asynccnt drain suffices; no workgroup barrier needed.
    asm volatile("s_wait_asynccnt 0" ::: "memory");

    // ---- LDS -> global: 128-bit async DMA stores, non-temporal (streaming,
    // never re-read; keep L2 free for weight-row reuse) ----
    asm volatile(
        "global_store_async_from_lds_b128 %2, %0, %4 th:TH_STORE_NT\n\t"
        "global_store_async_from_lds_b128 %2, %0, %4 offset:16 th:TH_STORE_NT\n\t"
        "global_store_async_from_lds_b128 %3, %1, %4 th:TH_STORE_NT\n\t"
        "global_store_async_from_lds_b128 %3, %1, %4 offset:16 th:TH_STORE_NT"
        :
        : "v"(lds0), "v"(lds1), "v"(dst0), "v"(dst1), "s"(out)
        : "memory");
    // S_ENDPGM performs an implicit wait-idle, draining ASYNCcnt before the
    // wave retires, so the stores are guaranteed to land.
}

extern "C" void kernel_launch(void* const* d_in, const int* in_sizes, int n_in,
                              void* d_out, int out_size, void* d_ws, size_t ws_size,
                              hipStream_t stream) {
    (void)n_in; (void)d_ws; (void)ws_size; (void)out_size;

    const int*   ids    = (const int*)d_in[0];    // [8*4096] int32 token ids
    const float* weight = (const float*)d_in[1];  // [50257*2048] f32 table
    float*       out    = (float*)d_out;          // [8*4096*2048] f32

    const int n_tok = in_sizes[0];                // 32768 tokens

    // Two rows per block; 16 KB of dynamic LDS reserves the async staging
    // buffer (offsets [0, 16384) are in-range for the DMA ops).
    const int n_blocks = (n_tok + ROWS_PER_BLOCK - 1) / ROWS_PER_BLOCK;
    dim3 grid((unsigned)n_blocks), block(256);
    embed_gather_async<<<grid, block, ROWS_PER_BLOCK * ROW_BYTES, stream>>>(
        ids, weight, out, n_tok);
}